// KalmanFilter_21792664060161
// MI455X (gfx1250) — compile-verified
//
#include <hip/hip_runtime.h>
#include <math.h>

// Kalman filter: G=256 chains, T=512 sequential steps, D=32 state, M=16 obs.
// One wave32 per group; cov/state live in LDS; all matrix products via
// V_WMMA_F32_16X16X4_F32 (fp32 WMMA, the correct precision for covariance math).

static constexpr int KG = 256;
static constexpr int KT = 512;
static constexpr int KM = 16;
static constexpr int KD = 32;

typedef __attribute__((ext_vector_type(2))) float v2f;
typedef __attribute__((ext_vector_type(8))) float v8f;

// ---- WMMA helpers -----------------------------------------------------------
// 16x16x16 tile product accumulated via 4x V_WMMA_F32_16X16X4_F32.
// A layout (16x4 f32): lanes 0-15 rows M=0..15 holding K={0,1}; lanes 16-31 K={2,3}.
// B layout (4x16 f32): VGPR0 = rows K={0 | 2} striped over lane halves, VGPR1 = K={1 | 3}.
// C/D layout (16x16 f32): VGPR e holds row e (lanes 0-15) / row e+8 (lanes 16-31).

__device__ __forceinline__ v8f mm_AB(v8f c,
                                     const float* A, int lda, int ar, int ac,
                                     const float* B, int ldb, int br, int bc) {
  const int l = threadIdx.x & 31;
  const int m = l & 15;
  const int half2 = (l >> 4) << 1;   // 0 or 2
#pragma unroll
  for (int kc = 0; kc < 4; ++kc) {
    v2f a, b;
    const int ka = ac + 4 * kc + half2;
    a.x = A[(ar + m) * lda + ka];
    a.y = A[(ar + m) * lda + ka + 1];
    const int kb = br + 4 * kc + half2;
    b.x = B[kb * ldb + bc + m];
    b.y = B[(kb + 1) * ldb + bc + m];
    c = __builtin_amdgcn_wmma_f32_16x16x4_f32(false, a, false, b, (short)0, c,
                                              false, false);
  }
  return c;
}

// Same, but the B operand is (Bt)^T: element(k,n) = Bt[(bc+n)*ldb + br+k].
// br/bc are coordinates in the *transposed* matrix. Used for H^T and F^T.
__device__ __forceinline__ v8f mm_ABt(v8f c,
                                      const float* A, int lda, int ar, int ac,
                                      const float* Bt, int ldb, int br, int bc) {
  const int l = threadIdx.x & 31;
  const int m = l & 15;
  const int half2 = (l >> 4) << 1;
#pragma unroll
  for (int kc = 0; kc < 4; ++kc) {
    v2f a, b;
    const int ka = ac + 4 * kc + half2;
    a.x = A[(ar + m) * lda + ka];
    a.y = A[(ar + m) * lda + ka + 1];
    const int kb = br + 4 * kc + half2;
    b.x = Bt[(bc + m) * ldb + kb];       // contiguous pair per lane
    b.y = Bt[(bc + m) * ldb + kb + 1];
    c = __builtin_amdgcn_wmma_f32_16x16x4_f32(false, a, false, b, (short)0, c,
                                              false, false);
  }
  return c;
}

__device__ __forceinline__ v8f loadC(const float* Mt, int ld, int r0, int c0) {
  const int l = threadIdx.x & 31;
  const int n = l & 15;
  const int half8 = (l >> 4) << 3;
  v8f c;
#pragma unroll
  for (int e = 0; e < 8; ++e) c[e] = Mt[(r0 + e + half8) * ld + c0 + n];
  return c;
}

__device__ __forceinline__ void storeC(float* Mt, int ld, int r0, int c0, v8f c) {
  const int l = threadIdx.x & 31;
  const int n = l & 15;
  const int half8 = (l >> 4) << 3;
#pragma unroll
  for (int e = 0; e < 8; ++e) Mt[(r0 + e + half8) * ld + c0 + n] = c[e];
}

__device__ __forceinline__ void storeCneg(float* Mt, int ld, int r0, int c0, v8f c) {
  const int l = threadIdx.x & 31;
  const int n = l & 15;
  const int half8 = (l >> 4) << 3;
#pragma unroll
  for (int e = 0; e < 8; ++e) Mt[(r0 + e + half8) * ld + c0 + n] = -c[e];
}

// Store a C-layout tile straight to global memory (row-major, leading dim ld).
__device__ __forceinline__ void storeC_global(float* __restrict__ Mt, int ld,
                                              int r0, int c0, v8f c) {
  const int l = threadIdx.x & 31;
  const int n = l & 15;
  const int half8 = (l >> 4) << 3;
#pragma unroll
  for (int e = 0; e < 8; ++e) Mt[(r0 + e + half8) * ld + c0 + n] = c[e];
}

// ---- kernel -----------------------------------------------------------------
__global__ __launch_bounds__(32) void KalmanFilter_gfx1250_kernel(
    const float* __restrict__ input,    // (G,T,M)
    const float* __restrict__ F_raw,    // (D,D)
    const float* __restrict__ Q_sqrt,   // (D,D)
    const float* __restrict__ H_raw,    // (M,D)
    const float* __restrict__ R_sqrt,   // (M,M)
    float* __restrict__ mean_out,       // (G,T,D)
    float* __restrict__ cov_out) {      // (G,T,D,D)
  __shared__ float sF[KD * KD];
  __shared__ float sH[KM * KD];
  __shared__ float sQ[KD * KD];
  __shared__ float sR[KM * KM];
  __shared__ float sCov[KD * KD];
  __shared__ float sTmp[KD * KD];       // staging + F*P
  __shared__ float sPHt[KD * KM];
  __shared__ float sHC[KM * KD];
  __shared__ float sKneg[KD * KM];      // holds -K
  __shared__ float sAug[KM * 2 * KM];   // [S | I] -> [I | S^-1]
  __shared__ float sMean[KD];
  __shared__ float sResid[KM];

  const int l = threadIdx.x;            // 0..31 (wave32)
  const int g = blockIdx.x;

  const float fscale = 0.5f / sqrtf((float)KD);
  const float hscale = 1.0f / sqrtf((float)KD);

  // --- build F, H, Q, R once per block (trivial vs. 512-step chain) ---
  for (int i = l; i < KD * KD; i += 32) sF[i] = fscale * F_raw[i];
  for (int i = l; i < KM * KD; i += 32) sH[i] = hscale * H_raw[i];
  for (int i = l; i < KD * KD; i += 32) sTmp[i] = Q_sqrt[i];
  __syncthreads();
  for (int i = l; i < KD * KD; i += 32) {
    const int r = i >> 5, c = i & 31;
    float s = 0.f;
    for (int k = 0; k < KD; ++k) s += sTmp[r * KD + k] * sTmp[c * KD + k];
    sQ[i] = (0.01f / (float)KD) * s + ((r == c) ? 1e-3f : 0.f);
  }
  for (int i = l; i < KM * KM; i += 32) sAug[i] = R_sqrt[i];
  __syncthreads();
  for (int i = l; i < KM * KM; i += 32) {
    const int r = i >> 4, c = i & 15;
    float s = 0.f;
    for (int k = 0; k < KM; ++k) s += sAug[r * KM + k] * sAug[c * KM + k];
    sR[i] = (0.1f / (float)KM) * s + ((r == c) ? 1e-3f : 0.f);
  }
  // cov0 = I, mean0 = 0
  for (int i = l; i < KD * KD; i += 32) sCov[i] = ((i >> 5) == (i & 31)) ? 1.f : 0.f;
  sMean[l] = 0.f;
  __syncthreads();

  const float* obsBase = input + (size_t)g * KT * KM;
  float* mOut = mean_out + (size_t)g * KT * KD;
  float* cOut = cov_out + (size_t)g * KT * KD * KD;

#pragma unroll 1
  for (int t = 0; t < KT; ++t) {
    const float* obs = obsBase + t * KM;

    // resid = obs - H @ mean
    if (l < KM) {
      float s = 0.f;
      for (int d = 0; d < KD; ++d) s += sH[l * KD + d] * sMean[d];
      sResid[l] = obs[l] - s;
    }
    __syncthreads();

    // PHt = P @ H^T   (32x16)
#pragma unroll
    for (int i = 0; i < 2; ++i) {
      v8f c = {};
#pragma unroll
      for (int k = 0; k < 2; ++k)
        c = mm_ABt(c, sCov, KD, 16 * i, 16 * k, sH, KD, 16 * k, 0);
      storeC(sPHt, KM, 16 * i, 0, c);
    }
    __syncthreads();

    // S = H @ PHt + R  -> left half of sAug; right half = I
    {
      v8f c = loadC(sR, KM, 0, 0);
#pragma unroll
      for (int k = 0; k < 2; ++k)
        c = mm_AB(c, sH, KD, 0, 16 * k, sPHt, KM, 16 * k, 0);
      storeC(sAug, 2 * KM, 0, 0, c);
    }
    for (int i = l; i < KM * KM; i += 32) {
      const int r = i >> 4, cc = i & 15;
      sAug[r * 2 * KM + KM + cc] = (r == cc) ? 1.f : 0.f;
    }
    __syncthreads();

    // Gauss-Jordan: [S | I] -> [I | S^-1]  (lane l owns column l)
#pragma unroll 1
    for (int p = 0; p < KM; ++p) {
      const float inv = 1.f / sAug[p * 2 * KM + p];
      sAug[p * 2 * KM + l] *= inv;
#pragma unroll 1
      for (int r = 0; r < KM; ++r) {
        if (r != p) {
          const float f = sAug[r * 2 * KM + p];
          sAug[r * 2 * KM + l] -= f * sAug[p * 2 * KM + l];
        }
      }
    }
    __syncthreads();

    // K = PHt @ S^-1 ; store -K
#pragma unroll
    for (int i = 0; i < 2; ++i) {
      v8f c = {};
      c = mm_AB(c, sPHt, KM, 16 * i, 0, sAug, 2 * KM, 0, KM);
      storeCneg(sKneg, KM, 16 * i, 0, c);
    }
    __syncthreads();

    // mean += K @ resid   (== mean - Kneg @ resid)
    {
      float s = 0.f;
      for (int m = 0; m < KM; ++m) s += sKneg[l * KM + m] * sResid[m];
      sMean[l] -= s;
    }

    // HC = H @ P   (16x32)
#pragma unroll
    for (int j = 0; j < 2; ++j) {
      v8f c = {};
#pragma unroll
      for (int k = 0; k < 2; ++k)
        c = mm_AB(c, sH, KD, 0, 16 * k, sCov, KD, 16 * k, 16 * j);
      storeC(sHC, KD, 0, 16 * j, c);
    }
    __syncthreads();

    // P = P + (-K) @ HC  == (I - K H) P
#pragma unroll
    for (int i = 0; i < 2; ++i)
#pragma unroll
      for (int j = 0; j < 2; ++j) {
        v8f c = loadC(sCov, KD, 16 * i, 16 * j);
        c = mm_AB(c, sKneg, KM, 16 * i, 0, sHC, KD, 0, 16 * j);
        storeC(sCov, KD, 16 * i, 16 * j, c);
      }
    __syncthreads();

    // Tmp = F @ P
#pragma unroll
    for (int i = 0; i < 2; ++i)
#pragma unroll
      for (int j = 0; j < 2; ++j) {
        v8f c = {};
#pragma unroll
        for (int k = 0; k < 2; ++k)
          c = mm_AB(c, sF, KD, 16 * i, 16 * k, sCov, KD, 16 * k, 16 * j);
        storeC(sTmp, KD, 16 * i, 16 * j, c);
      }
    __syncthreads();

    // P = Tmp @ F^T + Q  -> LDS (for next step) AND straight to global output
    float* ct = cOut + (size_t)t * KD * KD;
#pragma unroll
    for (int i = 0; i < 2; ++i)
#pragma unroll
      for (int j = 0; j < 2; ++j) {
        v8f c = loadC(sQ, KD, 16 * i, 16 * j);
#pragma unroll
        for (int k = 0; k < 2; ++k)
          c = mm_ABt(c, sTmp, KD, 16 * i, 16 * k, sF, KD, 16 * k, 16 * j);
        storeC(sCov, KD, 16 * i, 16 * j, c);
        storeC_global(ct, KD, 16 * i, 16 * j, c);   // no LDS round-trip for output
      }

    // mean = F @ mean
    float nm = 0.f;
    for (int k = 0; k < KD; ++k) nm += sF[l * KD + k] * sMean[k];
    __syncthreads();
    sMean[l] = nm;
    __syncthreads();

    // emit post-predict mean
    mOut[(size_t)t * KD + l] = nm;
  }
}

extern "C" void kernel_launch(void* const* d_in, const int* in_sizes, int n_in,
                              void* d_out, int out_size, void* d_ws, size_t ws_size,
                              hipStream_t stream) {
  (void)in_sizes; (void)n_in; (void)out_size; (void)d_ws; (void)ws_size;
  const float* input  = (const float*)d_in[0];
  const float* F_raw  = (const float*)d_in[1];
  const float* Q_sqrt = (const float*)d_in[2];
  const float* H_raw  = (const float*)d_in[3];
  const float* R_sqrt = (const float*)d_in[4];
  float* mean_out = (float*)d_out;                       // (G,T,D,1) flat
  float* cov_out  = mean_out + (size_t)KG * KT * KD;     // (G,T,D,D) flat

  KalmanFilter_gfx1250_kernel<<<dim3(KG), dim3(32), 0, stream>>>(
      input, F_raw, Q_sqrt, H_raw, R_sqrt, mean_out, cov_out);
}